// Joint_50766513439136
// MI455X (gfx1250) — compile-verified
//
#include <hip/hip_runtime.h>
#include <hip/hip_bf16.h>
#include <stdint.h>
#include <math.h>

// ---------------------------------------------------------------------------
// Sizes from the reference
// ---------------------------------------------------------------------------
#define Bn   16
#define En   64
#define Sn   256
#define UPn  512
#define Mn   6
#define Rn   60
#define Dn   120                 // 2*R
#define DOT2 100                 // DOT=int(4*512/200)=10 -> 100
#define SS   (Sn * Sn)           // 65536
#define UU   (UPn * UPn)         // 262144

typedef __attribute__((ext_vector_type(2))) float v2f;
typedef __attribute__((ext_vector_type(8))) float v8f;
typedef unsigned int v4u __attribute__((ext_vector_type(4)));
typedef int          v4i __attribute__((ext_vector_type(4)));
typedef int          v8i __attribute__((ext_vector_type(8)));

// ---------------------------------------------------------------------------
// 1) Decoder GEMM + sigmoid via V_WMMA_F32_16X16X4_F32
//    pred[b,s] = sigmoid( sum_k kout[b,k]*wdec[k,s] + bdec[s] )
//    One wave handles a 16(b) x 16(s) tile; K=64 -> 16 WMMA steps of K=4.
// ---------------------------------------------------------------------------
__global__ void __launch_bounds__(128)
k_gemm_sigmoid(const float* __restrict__ kout, const float* __restrict__ wdec,
               const float* __restrict__ bdec, float* __restrict__ pred) {
  const int lane = threadIdx.x & 31;
  const int wave = threadIdx.x >> 5;
  const int tile = blockIdx.x * 4 + wave;      // 0..4095
  const int s0   = tile * 16;
  const int nm   = lane & 15;                  // M for A, N for B/D
  const int hk   = lane >> 4;                  // K half-wave select
  v8f acc = {0.f, 0.f, 0.f, 0.f, 0.f, 0.f, 0.f, 0.f};
  for (int kc = 0; kc < 16; ++kc) {
    const int k = kc * 4 + hk * 2;
    v2f a, b;
    a.x = kout[nm * En + k];
    a.y = kout[nm * En + k + 1];
    b.x = wdec[(size_t)(k + 0) * SS + s0 + nm];
    b.y = wdec[(size_t)(k + 1) * SS + s0 + nm];
    acc = __builtin_amdgcn_wmma_f32_16x16x4_f32(false, a, false, b,
                                                (short)0, acc, false, false);
  }
  const float bias = bdec[s0 + nm];
#pragma unroll
  for (int r = 0; r < 8; ++r) {
    const int bidx = r + 8 * hk;
    const float v  = acc[r] + bias;
    pred[(size_t)bidx * SS + s0 + nm] = 1.f / (1.f + expf(-v));
  }
}

// ---------------------------------------------------------------------------
// 2) Per-batch inverse 2x3 affines (16 threads)
// ---------------------------------------------------------------------------
__device__ __forceinline__ void inv2x3(float a, float b, float c,
                                       float d, float e, float f, float* o) {
  const float det = a * e - b * d;
  const float id  = 1.f / det;
  o[0] =  e * id; o[1] = -b * id; o[2] = (b * f - c * e) * id;
  o[3] = -d * id; o[4] =  a * id; o[5] = (c * d - a * f) * id;
}

__global__ void k_inv(const float* __restrict__ angle, const float* __restrict__ scale,
                      const float* __restrict__ shear,
                      float* __restrict__ inv1, float* __restrict__ inv2) {
  const int b = threadIdx.x;
  if (b >= Bn) return;
  const float c = cosf(angle[b]), s = sinf(angle[b]);
  inv2x3(c, -s, 0.f, s, c, 0.f, inv2 + b * 6);
  inv2x3(scale[b * 2 + 0], shear[b], 0.f, 0.f, scale[b * 2 + 1], 0.f, inv1 + b * 6);
}

// ---------------------------------------------------------------------------
// 3) 2x bilinear upsample 256 -> 512 (half-pixel centers, edge clamp)
// ---------------------------------------------------------------------------
__global__ void k_resize2x(const float* __restrict__ in, float* __restrict__ out) {
  const int idx = blockIdx.x * blockDim.x + threadIdx.x;
  if (idx >= Bn * UU) return;
  const int j = idx & (UPn - 1);
  const int i = (idx >> 9) & (UPn - 1);
  const int b = idx >> 18;
  const float fy = (i + 0.5f) * 0.5f - 0.5f;
  const float fx = (j + 0.5f) * 0.5f - 0.5f;
  const float y0f = floorf(fy), x0f = floorf(fx);
  const float wy1 = fy - y0f,  wx1 = fx - x0f;
  const int y0 = min(max((int)y0f, 0), Sn - 1);
  const int y1 = min(max((int)y0f + 1, 0), Sn - 1);
  const int x0 = min(max((int)x0f, 0), Sn - 1);
  const int x1 = min(max((int)x0f + 1, 0), Sn - 1);
  const float* p = in + (size_t)b * SS;
  const float v00 = p[y0 * Sn + x0], v01 = p[y0 * Sn + x1];
  const float v10 = p[y1 * Sn + x0], v11 = p[y1 * Sn + x1];
  out[idx] = (1.f - wy1) * ((1.f - wx1) * v00 + wx1 * v01) +
             wy1        * ((1.f - wx1) * v10 + wx1 * v11);
}

// ---------------------------------------------------------------------------
// grid_sample helpers: 4-corner bilinear with zero padding
// ---------------------------------------------------------------------------
__device__ __forceinline__ float sample_zero_f32(const float* __restrict__ img,
                                                 int H, int W, float px, float py) {
  const float x0f = floorf(px), y0f = floorf(py);
  const int   x0 = (int)x0f,    y0 = (int)y0f;
  const float wx1 = px - x0f,   wy1 = py - y0f;
  const float wx0 = 1.f - wx1,  wy0 = 1.f - wy1;
  float acc = 0.f;
  if (x0 >= 0     && x0 < W     && y0 >= 0     && y0 < H)     acc += img[y0 * W + x0]           * wx0 * wy0;
  if (x0 + 1 >= 0 && x0 + 1 < W && y0 >= 0     && y0 < H)     acc += img[y0 * W + x0 + 1]       * wx1 * wy0;
  if (x0 >= 0     && x0 < W     && y0 + 1 >= 0 && y0 + 1 < H) acc += img[(y0 + 1) * W + x0]     * wx0 * wy1;
  if (x0 + 1 >= 0 && x0 + 1 < W && y0 + 1 >= 0 && y0 + 1 < H) acc += img[(y0 + 1) * W + x0 + 1] * wx1 * wy1;
  return acc;
}

__device__ __forceinline__ float sample_zero_u8(const uint8_t* __restrict__ img,
                                                int H, int W, float px, float py) {
  const float x0f = floorf(px), y0f = floorf(py);
  const int   x0 = (int)x0f,    y0 = (int)y0f;
  const float wx1 = px - x0f,   wy1 = py - y0f;
  const float wx0 = 1.f - wx1,  wy0 = 1.f - wy1;
  float acc = 0.f;
  if (x0 >= 0     && x0 < W     && y0 >= 0     && y0 < H)     acc += (float)img[y0 * W + x0]           * wx0 * wy0;
  if (x0 + 1 >= 0 && x0 + 1 < W && y0 >= 0     && y0 < H)     acc += (float)img[y0 * W + x0 + 1]       * wx1 * wy0;
  if (x0 >= 0     && x0 < W     && y0 + 1 >= 0 && y0 + 1 < H) acc += (float)img[(y0 + 1) * W + x0]     * wx0 * wy1;
  if (x0 + 1 >= 0 && x0 + 1 < W && y0 + 1 >= 0 && y0 + 1 < H) acc += (float)img[(y0 + 1) * W + x0 + 1] * wx1 * wy1;
  return acc;
}

__device__ __forceinline__ void warp_coords(int i, int j, const float* t,
                                            float& px, float& py) {
  const float xn = (j + 0.5f) * (2.f / UPn) - 1.f;
  const float yn = (i + 0.5f) * (2.f / UPn) - 1.f;
  const float gx = t[0] * xn + t[1] * yn + t[2];
  const float gy = t[3] * xn + t[4] * yn + t[5];
  px = (gx + 1.f) * (UPn * 0.5f) - 0.5f;   // indexes W (axis 1)
  py = (gy + 1.f) * (UPn * 0.5f) - 0.5f;   // indexes H (axis 0)
}

// 4) affine warp f32 -> f32, per-batch input planes
__global__ void k_warp_f32(const float* __restrict__ in, float* __restrict__ out,
                           const float* __restrict__ theta) {
  const int idx = blockIdx.x * blockDim.x + threadIdx.x;
  if (idx >= Bn * UU) return;
  const int j = idx & (UPn - 1), i = (idx >> 9) & (UPn - 1), b = idx >> 18;
  float px, py;
  warp_coords(i, j, theta + b * 6, px, py);
  out[idx] = sample_zero_f32(in + (size_t)b * UU, UPn, UPn, px, py);
}

// 5a) warp of one broadcast uint8 mask plane -> per-batch f32
__global__ void k_warp_mask1(const uint8_t* __restrict__ mask, float* __restrict__ out,
                             const float* __restrict__ theta) {
  const int idx = blockIdx.x * blockDim.x + threadIdx.x;
  if (idx >= Bn * UU) return;
  const int j = idx & (UPn - 1), i = (idx >> 9) & (UPn - 1), b = idx >> 18;
  float px, py;
  warp_coords(i, j, theta + b * 6, px, py);
  out[idx] = sample_zero_u8(mask, UPn, UPn, px, py);
}

// 5b) warp per-batch f32 -> thresholded uint8 stored at (b*Mn + m) plane
__global__ void k_warp_mask2(const float* __restrict__ in, uint8_t* __restrict__ out_m,
                             const float* __restrict__ theta) {
  const int idx = blockIdx.x * blockDim.x + threadIdx.x;
  if (idx >= Bn * UU) return;
  const int j = idx & (UPn - 1), i = (idx >> 9) & (UPn - 1), b = idx >> 18;
  float px, py;
  warp_coords(i, j, theta + b * 6, px, py);
  const float v = sample_zero_f32(in + (size_t)b * UU, UPn, UPn, px, py);
  out_m[(size_t)b * (Mn * UU) + i * UPn + j] = (v >= 0.5f) ? 1 : 0;
}

// ---------------------------------------------------------------------------
// 6) Reductions: scan-step slice origins depend only on `orig` + masks.
// ---------------------------------------------------------------------------
__global__ void __launch_bounds__(256)
k_reduce1(const uint8_t* __restrict__ masks, const float* __restrict__ orig,
          float* __restrict__ red) {
  const int blk = blockIdx.x;            // b*Mn + m
  const int b   = blk / Mn;
  const uint8_t* mk = masks + (size_t)blk * UU;
  const float*   og = orig  + (size_t)b  * UU;
  float s0 = 0.f, s1 = 0.f, s5 = 0.f, s6 = 0.f;
  for (int t = threadIdx.x; t < UU; t += 256) {
    const float mv = (float)mk[t];
    const int i = t >> 9, j = t & (UPn - 1);
    s0 += mv;
    s1 += og[t] * mv;
    s5 += (float)i * mv;
    s6 += (float)j * mv;
  }
  __shared__ float sh[4][256];
  sh[0][threadIdx.x] = s0; sh[1][threadIdx.x] = s1;
  sh[2][threadIdx.x] = s5; sh[3][threadIdx.x] = s6;
  __syncthreads();
  for (int off = 128; off > 0; off >>= 1) {
    if (threadIdx.x < off)
      for (int q = 0; q < 4; ++q)
        sh[q][threadIdx.x] += sh[q][threadIdx.x + off];
    __syncthreads();
  }
  if (threadIdx.x == 0) {
    const float cnt = fmaxf(sh[0][0], 1.f);
    red[blk * 4 + 0] = cnt;
    red[blk * 4 + 1] = sh[1][0] / cnt;   // mean_mass
    red[blk * 4 + 2] = sh[2][0];         // sum(rows*m2d)
    red[blk * 4 + 3] = sh[3][0];         // sum(cols*m2d)
  }
}

__global__ void __launch_bounds__(256)
k_reduce2(const uint8_t* __restrict__ masks, const float* __restrict__ orig,
          const float* __restrict__ red, int* __restrict__ sxsy) {
  const int blk = blockIdx.x;
  const int b   = blk / Mn;
  const uint8_t* mk = masks + (size_t)blk * UU;
  const float*   og = orig  + (size_t)b  * UU;
  const float mean = red[blk * 4 + 1];
  float s2 = 0.f, s3 = 0.f, s4 = 0.f;
  for (int t = threadIdx.x; t < UU; t += 256) {
    const float mv = (float)mk[t];
    const float mass = fmaxf(og[t] - 1.5f * mean, 0.f) * mv;
    const int i = t >> 9, j = t & (UPn - 1);
    s2 += mass;
    s3 += (float)i * mass;
    s4 += (float)j * mass;
  }
  __shared__ float sh[3][256];
  sh[0][threadIdx.x] = s2; sh[1][threadIdx.x] = s3; sh[2][threadIdx.x] = s4;
  __syncthreads();
  for (int off = 128; off > 0; off >>= 1) {
    if (threadIdx.x < off)
      for (int q = 0; q < 3; ++q)
        sh[q][threadIdx.x] += sh[q][threadIdx.x + off];
    __syncthreads();
  }
  if (threadIdx.x == 0) {
    const float cnt = red[blk * 4 + 0];
    const float sm  = sh[0][0];
    float cx, cy;
    if (sm > 0.f) { cx = sh[1][0] / sm;  cy = sh[2][0] / sm; }
    else          { cx = red[blk * 4 + 2] / cnt; cy = red[blk * 4 + 3] / cnt; }
    sxsy[blk * 2 + 0] = (int)fminf(fmaxf(rintf(cx) - (float)Rn, 0.f), (float)(UPn - Dn));
    sxsy[blk * 2 + 1] = (int)fminf(fmaxf(rintf(cy) - (float)Rn, 0.f), (float)(UPn - Dn));
  }
}

// ---------------------------------------------------------------------------
// 7) Scan step m: TDM-stage the 120x120 tile into LDS (tensor_load_to_lds),
//    wait TENSORcnt, disc-divide in LDS, barrier, resample with inv1 grid,
//    write back in place.
//    D# group0: count=1 | lds_addr | global_addr(57b) | type=2
//    D# group1: data_size=4B, tile=120x120, tensor dims 2^30 (never OOB --
//               slice origin already clipped to <=392), dim0 stride 512.
// ---------------------------------------------------------------------------
__global__ void __launch_bounds__(256)
k_revise(float* __restrict__ img, const int* __restrict__ sxsy,
         const float* __restrict__ adj, const float* __restrict__ inv1, int m) {
  __shared__ float tile[Dn * Dn];        // 57.6 KB
  const int b  = blockIdx.x;
  const int sx = sxsy[(b * Mn + m) * 2 + 0];
  const int sy = sxsy[(b * Mn + m) * 2 + 1];
  const float ad = adj[b];
  const float* t = inv1 + b * 6;
  const float t0 = t[0], t1 = t[1], t2 = t[2], t3 = t[3], t4 = t[4], t5 = t[5];
  float* ib = img + (size_t)b * UU;
  const float* tsrc = ib + (size_t)sx * UPn + sy;

  if ((threadIdx.x >> 5) == 0) {         // one wave issues the TDM op
    const unsigned long long ga = (unsigned long long)(uintptr_t)tsrc;
    const unsigned int lds_off = (unsigned int)(uintptr_t)tile;  // LDS byte offset
    v4u g0 = { 1u,                                       // count=1, flags 0
               lds_off,                                  // lds_addr
               (unsigned int)(ga & 0xFFFFFFFFu),         // global_addr[31:0]
               (unsigned int)((ga >> 32) & 0x1FFFFFFu) | (2u << 30) }; // [56:32]|type=2
    v8i g1 = { 0x00020000,                // wg_mask=0, data_size=2 (4 bytes)
               0,                         // atomic_barrier=0; tensor_dim0[15:0]=0
               0x00004000,                // tensor_dim0[31:16]=2^30>>16; tensor_dim1[15:0]=0
               0x00004000 | (Dn << 16),   // tensor_dim1[31:16]; tile_dim0=120
               Dn,                        // tile_dim1=120; tile_dim2=0
               UPn,                       // tensor_dim0_stride[31:0]=512
               0,                         // stride0[47:32]=0; stride1[15:0]=0
               4 };                       // tensor_dim1_stride[47:16] -> 262144
    v4i g2 = {0, 0, 0, 0};
    v4i g3 = {0, 0, 0, 0};
#if defined(__clang_major__) && (__clang_major__ >= 23)
    v8i gpad = {0, 0, 0, 0, 0, 0, 0, 0};
    __builtin_amdgcn_tensor_load_to_lds(g0, g1, g2, g3, gpad, 0);
#else
    __builtin_amdgcn_tensor_load_to_lds(g0, g1, g2, g3, 0);
#endif
    __builtin_amdgcn_s_wait_tensorcnt(0);
  }
  __syncthreads();

  // disc division in LDS
  for (int p = threadIdx.x; p < Dn * Dn; p += 256) {
    const int i = p / Dn, j = p - i * Dn;
    const int di = i - Rn, dj = j - Rn;
    if (di * di + dj * dj <= DOT2) tile[p] /= ad;
  }
  __syncthreads();

  // resample with the inv1 affine grid (affine_grid(inv1, D, D)) and write back
  for (int p = threadIdx.x; p < Dn * Dn; p += 256) {
    const int i = p / Dn, j = p - i * Dn;
    const float xn = (j + 0.5f) * (2.f / Dn) - 1.f;
    const float yn = (i + 0.5f) * (2.f / Dn) - 1.f;
    const float gx = t0 * xn + t1 * yn + t2;
    const float gy = t3 * xn + t4 * yn + t5;
    const float px = (gx + 1.f) * (Dn * 0.5f) - 0.5f;
    const float py = (gy + 1.f) * (Dn * 0.5f) - 0.5f;
    ib[(size_t)(sx + i) * UPn + (sy + j)] = sample_zero_f32(tile, Dn, Dn, px, py);
  }
}

// ---------------------------------------------------------------------------
// Orchestration. Workspace layout (~76 MB):
//   [0,4M)    pred_base  (16 x 65536 f32)
//   [4M,20M)  bufA       (16 x 512 x 512 f32)
//   [20M,36M) bufB       (16 x 512 x 512 f32)
//   [36M,52M) orig       (16 x 512 x 512 f32)
//   [52M,76M) masks_new  (16 x 6 x 512 x 512 u8)
//   [76M,..)  inv1/inv2/red/sxsy scalars
// d_out is the evolving image (scan carry).
// ---------------------------------------------------------------------------
extern "C" void kernel_launch(void* const* d_in, const int* in_sizes, int n_in,
                              void* d_out, int out_size, void* d_ws, size_t ws_size,
                              hipStream_t stream) {
  (void)in_sizes; (void)n_in; (void)out_size; (void)ws_size;
  const float*   kout     = (const float*)d_in[1];
  const float*   wdec     = (const float*)d_in[2];
  const float*   bdec     = (const float*)d_in[3];
  const float*   angle    = (const float*)d_in[4];
  const float*   scale    = (const float*)d_in[5];
  const float*   shear    = (const float*)d_in[6];
  const float*   adj      = (const float*)d_in[7];
  const uint8_t* masks_in = (const uint8_t*)d_in[8];   // jnp bool -> 1 byte/elem
  float* out = (float*)d_out;

  char* ws = (char*)d_ws;
  float*   pred_base = (float*)(ws);
  float*   bufA      = (float*)(ws + ((size_t)4  << 20));
  float*   bufB      = (float*)(ws + ((size_t)20 << 20));
  float*   orig      = (float*)(ws + ((size_t)36 << 20));
  uint8_t* masks_new = (uint8_t*)(ws + ((size_t)52 << 20));
  float*   scal      = (float*)(ws + ((size_t)76 << 20));
  float*   inv1 = scal;                 // 96 f32
  float*   inv2 = scal + 96;            // 96 f32
  float*   red  = scal + 192;           // 96 * 4 f32
  int*     sxsy = (int*)(scal + 576);   // 96 * 2 i32

  const int NPIX   = Bn * UU;
  const int BLK    = 256;
  const int GRIDPX = (NPIX + BLK - 1) / BLK;

  // 1) decoder GEMM + sigmoid (WMMA f32 16x16x4)
  k_gemm_sigmoid<<<1024, 128, 0, stream>>>(kout, wdec, bdec, pred_base);
  // 2) per-batch inverse affines
  k_inv<<<1, 16, 0, stream>>>(angle, scale, shear, inv1, inv2);
  // 3) bilinear upsample 256 -> 512
  k_resize2x<<<GRIDPX, BLK, 0, stream>>>(pred_base, bufA);
  // 4) pred_rot = warp(pred_base_inp, inv2); pred_inp = warp(pred_rot, inv1)
  k_warp_f32<<<GRIDPX, BLK, 0, stream>>>(bufA, bufB, inv2);
  k_warp_f32<<<GRIDPX, BLK, 0, stream>>>(bufB, orig, inv1);
  // scan carry starts as orig
  hipMemcpyAsync(out, orig, (size_t)NPIX * sizeof(float),
                 hipMemcpyDeviceToDevice, stream);
  // 5) masks: warp twice, threshold; loop over m keeps temp at 16 MB
  for (int m = 0; m < Mn; ++m) {
    k_warp_mask1<<<GRIDPX, BLK, 0, stream>>>(masks_in + (size_t)m * UU, bufA, inv2);
    k_warp_mask2<<<GRIDPX, BLK, 0, stream>>>(bufA, masks_new + (size_t)m * UU, inv1);
  }
  // 6) precompute all 96 (b,m) slice origins
  k_reduce1<<<Bn * Mn, 256, 0, stream>>>(masks_new, orig, red);
  k_reduce2<<<Bn * Mn, 256, 0, stream>>>(masks_new, orig, red, sxsy);
  // 7) sequential scan over masks; per-batch parallel (16 blocks each)
  for (int m = 0; m < Mn; ++m)
    k_revise<<<Bn, 256, 0, stream>>>(out, sxsy, adj, inv1, m);
}